// Final_Verfication_Transformer_6906307412227
// MI455X (gfx1250) — compile-verified
//
#include <hip/hip_runtime.h>

#define NN    2048
#define MPAD  2048
#define MREAL 2047
#define NHIST 10
#define KB    32
#define BTM   128
#define BTN   128
#define BS    36          // padded LDS stride for transposed B tiles

typedef __attribute__((ext_vector_type(2))) float v2f;
typedef __attribute__((ext_vector_type(4))) float v4f;
typedef __attribute__((ext_vector_type(8))) float v8f;

// ---------------------------------------------------------------- init carry
__global__ void init_carry(const float* __restrict__ A, const float* __restrict__ b,
                           float* __restrict__ Al, float* __restrict__ Au,
                           float* __restrict__ bl, float* __restrict__ bu) {
    int idx = blockIdx.x * blockDim.x + threadIdx.x;      // over MPAD*NN
    int row = idx / NN;
    float v = (row < MREAL) ? A[idx] : 0.0f;
    Al[idx] = v;
    Au[idx] = v;
    if (idx < MPAD) {
        float bb = (idx < MREAL) ? b[idx] : 0.0f;
        bl[idx] = bb;
        bu[idx] = bb;
    }
}

// ------------------------------------------------- fused Al/Au backsub GEMM
// new_Al = max(Al,0) @ dAl + min(Al,0) @ dAu
// new_Au = max(Au,0) @ dAu + min(Au,0) @ dAl
__global__ __launch_bounds__(256, 1)
void step_mat(const float* __restrict__ Alc, const float* __restrict__ Auc,
              const float* __restrict__ dAl, const float* __restrict__ dAu,
              float* __restrict__ Alo, float* __restrict__ Auo) {
    __shared__ float sAlp[BTM * KB];
    __shared__ float sAln[BTM * KB];
    __shared__ float sAup[BTM * KB];
    __shared__ float sAun[BTM * KB];
    __shared__ float sBl[BTN * BS];   // transposed: [n][k]
    __shared__ float sBu[BTN * BS];

    const int tid   = threadIdx.x;
    const int lane  = tid & 31;
    const int w     = tid >> 5;          // 0..7
    const int g     = lane >> 4;         // half-wave
    const int r     = lane & 15;
    const int mP    = (w & 3) * 32;      // wave's 2 m-stripes: mP + {0,16}
    const int nQ    = (w >> 2) * 64;     // wave's 4 n-tiles:  nQ + {0,16,32,48}
    const int blockM = blockIdx.y * BTM;
    const int blockN = blockIdx.x * BTN;

    v8f accl[2][4];
    v8f accu[2][4];
    #pragma unroll
    for (int s = 0; s < 2; ++s)
        #pragma unroll
        for (int t = 0; t < 4; ++t) { accl[s][t] = v8f{}; accu[s][t] = v8f{}; }

    // A loader: 128x32 tile per matrix, 256 threads -> 4 passes of v4f
    const int arow = tid >> 3;           // 0..31
    const int acol = (tid & 7) * 4;      // 0..28
    // B loader: 32x128 tile per matrix, 256 threads -> 4 passes of v4f (stored transposed)
    const int brow = tid >> 5;           // 0..7
    const int bcol = (tid & 31) * 4;     // 0..124

    for (int k0 = 0; k0 < NN; k0 += KB) {
        // ---- stage carry tiles with ReLU split done once
        #pragma unroll
        for (int rr = 0; rr < BTM; rr += 32) {
            v4f a = *(const v4f*)&Alc[(size_t)(blockM + arow + rr) * NN + k0 + acol];
            v4f u = *(const v4f*)&Auc[(size_t)(blockM + arow + rr) * NN + k0 + acol];
            v4f ap, an, up, un;
            #pragma unroll
            for (int c = 0; c < 4; ++c) {
                ap[c] = fmaxf(a[c], 0.0f); an[c] = fminf(a[c], 0.0f);
                up[c] = fmaxf(u[c], 0.0f); un[c] = fminf(u[c], 0.0f);
            }
            int o = (arow + rr) * KB + acol;
            *(v4f*)&sAlp[o] = ap;  *(v4f*)&sAln[o] = an;
            *(v4f*)&sAup[o] = up;  *(v4f*)&sAun[o] = un;
        }
        // ---- stage hist tiles transposed so B fragments are contiguous b64
        #pragma unroll
        for (int rr = 0; rr < KB; rr += 8) {
            v4f bldat = *(const v4f*)&dAl[(size_t)(k0 + brow + rr) * NN + blockN + bcol];
            v4f budat = *(const v4f*)&dAu[(size_t)(k0 + brow + rr) * NN + blockN + bcol];
            #pragma unroll
            for (int c = 0; c < 4; ++c) {
                sBl[(bcol + c) * BS + brow + rr] = bldat[c];
                sBu[(bcol + c) * BS + brow + rr] = budat[c];
            }
        }
        if (k0 + KB < NN) {  // prefetch next hist chunk (global_prefetch_b8)
            __builtin_prefetch(&dAl[(size_t)(k0 + KB + brow) * NN + blockN + bcol], 0, 3);
            __builtin_prefetch(&dAu[(size_t)(k0 + KB + brow) * NN + blockN + bcol], 0, 3);
        }
        __syncthreads();

        // ---- 8 x (K=4) WMMA steps; 32 wmma per step per wave (16 ds loads)
        #pragma unroll
        for (int kk = 0; kk < KB; kk += 4) {
            v2f alp[2], aln[2], aup[2], aun[2];
            #pragma unroll
            for (int s = 0; s < 2; ++s) {
                const int ak = (mP + 16 * s + r) * KB + kk + 2 * g;  // lane(g,m=r): K=2g,2g+1
                alp[s] = *(const v2f*)&sAlp[ak];
                aln[s] = *(const v2f*)&sAln[ak];
                aup[s] = *(const v2f*)&sAup[ak];
                aun[s] = *(const v2f*)&sAun[ak];
            }
            v2f blf[4], buf[4];
            #pragma unroll
            for (int t = 0; t < 4; ++t) {
                const int bk = (nQ + 16 * t + r) * BS + kk + 2 * g;
                blf[t] = *(const v2f*)&sBl[bk];
                buf[t] = *(const v2f*)&sBu[bk];
            }
            #pragma unroll
            for (int s = 0; s < 2; ++s) {
                #pragma unroll
                for (int t = 0; t < 4; ++t) {
                    accl[s][t] = __builtin_amdgcn_wmma_f32_16x16x4_f32(false, alp[s], false, blf[t], (short)0, accl[s][t], false, false);
                    accl[s][t] = __builtin_amdgcn_wmma_f32_16x16x4_f32(false, aln[s], false, buf[t], (short)0, accl[s][t], false, false);
                    accu[s][t] = __builtin_amdgcn_wmma_f32_16x16x4_f32(false, aup[s], false, buf[t], (short)0, accu[s][t], false, false);
                    accu[s][t] = __builtin_amdgcn_wmma_f32_16x16x4_f32(false, aun[s], false, blf[t], (short)0, accu[s][t], false, false);
                }
            }
        }
        __syncthreads();
    }

    // ---- epilogue: C/D layout — VGPR i : M = i + 8g, N = r
    #pragma unroll
    for (int s = 0; s < 2; ++s) {
        #pragma unroll
        for (int t = 0; t < 4; ++t) {
            #pragma unroll
            for (int i = 0; i < 8; ++i) {
                size_t row = (size_t)(blockM + mP + 16 * s + i + 8 * g) * NN;
                Alo[row + blockN + nQ + 16 * t + r] = accl[s][t][i];
                Auo[row + blockN + nQ + 16 * t + r] = accu[s][t][i];
            }
        }
    }
}

// ---------------------------------------------------------------- bias step
__global__ void step_bias(const float* __restrict__ Alc, const float* __restrict__ Auc,
                          const float* __restrict__ dbl_, const float* __restrict__ dbu_,
                          const float* __restrict__ blc, const float* __restrict__ buc,
                          float* __restrict__ blo, float* __restrict__ buo) {
    __shared__ float sl[256], su[256];
    int m = blockIdx.x;
    float suml = 0.0f, sumu = 0.0f;
    for (int k = threadIdx.x; k < NN; k += 256) {
        float al = Alc[(size_t)m * NN + k];
        float au = Auc[(size_t)m * NN + k];
        float dl = dbl_[k], du = dbu_[k];
        suml += fmaxf(al, 0.0f) * dl + fminf(al, 0.0f) * du;
        sumu += fmaxf(au, 0.0f) * du + fminf(au, 0.0f) * dl;
    }
    sl[threadIdx.x] = suml; su[threadIdx.x] = sumu;
    __syncthreads();
    for (int s = 128; s > 0; s >>= 1) {
        if (threadIdx.x < s) {
            sl[threadIdx.x] += sl[threadIdx.x + s];
            su[threadIdx.x] += su[threadIdx.x + s];
        }
        __syncthreads();
    }
    if (threadIdx.x == 0) {
        blo[m] = sl[0] + blc[m];
        buo[m] = su[0] + buc[m];
    }
}

// ------------------------------------------------------------ concretization
__global__ void final_bounds(const float* __restrict__ Al, const float* __restrict__ Au,
                             const float* __restrict__ bl, const float* __restrict__ bu,
                             const float* __restrict__ lo1, const float* __restrict__ up1,
                             float* __restrict__ out) {
    __shared__ float sl[256], su[256];
    int m = blockIdx.x;                 // 0..MREAL-1
    float suml = 0.0f, sumu = 0.0f;
    for (int k = threadIdx.x; k < NN; k += 256) {
        float al = Al[(size_t)m * NN + k];
        float au = Au[(size_t)m * NN + k];
        float lo = lo1[k], up = up1[k];
        suml += fmaxf(al, 0.0f) * lo + fminf(al, 0.0f) * up;
        sumu += fmaxf(au, 0.0f) * up + fminf(au, 0.0f) * lo;
    }
    sl[threadIdx.x] = suml; su[threadIdx.x] = sumu;
    __syncthreads();
    for (int s = 128; s > 0; s >>= 1) {
        if (threadIdx.x < s) {
            sl[threadIdx.x] += sl[threadIdx.x + s];
            su[threadIdx.x] += su[threadIdx.x + s];
        }
        __syncthreads();
    }
    if (threadIdx.x == 0) {
        out[m]         = sl[0] + bl[m];
        out[MREAL + m] = su[0] + bu[m];
    }
}

// ----------------------------------------------------------------- launcher
extern "C" void kernel_launch(void* const* d_in, const int* in_sizes, int n_in,
                              void* d_out, int out_size, void* d_ws, size_t ws_size,
                              hipStream_t stream) {
    const float* hist_Al = (const float*)d_in[0];   // (10, 2048, 2048)
    const float* hist_Au = (const float*)d_in[1];
    const float* hist_bl = (const float*)d_in[2];   // (10, 2048)
    const float* hist_bu = (const float*)d_in[3];
    const float* lower1  = (const float*)d_in[4];   // (2048,)
    const float* upper1  = (const float*)d_in[5];
    const float* A       = (const float*)d_in[6];   // (2047, 2048)
    const float* b       = (const float*)d_in[7];   // (2047,)

    const size_t MAT = (size_t)MPAD * NN;           // 4M floats
    float* W = (float*)d_ws;
    float* Albuf[2] = { W,            W + 2 * MAT };
    float* Aubuf[2] = { W + MAT,      W + 3 * MAT };
    float* base     = W + 4 * MAT;
    float* blbuf[2] = { base,             base + 2 * MPAD };
    float* bubuf[2] = { base + MPAD,      base + 3 * MPAD };

    init_carry<<<(int)(MAT / 256), 256, 0, stream>>>(A, b, Albuf[0], Aubuf[0], blbuf[0], bubuf[0]);

    dim3 grid(NN / BTN, MPAD / BTM);   // 16 x 16
    int cur = 0;
    for (int t = NHIST - 1; t >= 0; --t) {
        const float* dAl = hist_Al + (size_t)t * NN * NN;
        const float* dAu = hist_Au + (size_t)t * NN * NN;
        step_mat<<<grid, 256, 0, stream>>>(Albuf[cur], Aubuf[cur], dAl, dAu,
                                           Albuf[cur ^ 1], Aubuf[cur ^ 1]);
        step_bias<<<MPAD, 256, 0, stream>>>(Albuf[cur], Aubuf[cur],
                                            hist_bl + (size_t)t * NN, hist_bu + (size_t)t * NN,
                                            blbuf[cur], bubuf[cur],
                                            blbuf[cur ^ 1], bubuf[cur ^ 1]);
        cur ^= 1;
    }

    final_bounds<<<MREAL, 256, 0, stream>>>(Albuf[cur], Aubuf[cur], blbuf[cur], bubuf[cur],
                                            lower1, upper1, (float*)d_out);
}